// Quantization_3401614099091
// MI455X (gfx1250) — compile-verified
//
#include <hip/hip_runtime.h>

typedef float v2f __attribute__((ext_vector_type(2)));
typedef float v8f __attribute__((ext_vector_type(8)));

#define N_TOTAL 8192
#define M_SUB   64
#define K_CB    256
#define D_SUB   16
#define VDIM    (M_SUB * D_SUB)   // 1024 floats per vector row
#define CBP     20                // padded LDS pitch (floats): 80B rows, 16B aligned, conflict-free
#define WPB     8                 // waves per block
#define TPW     4                 // 16-row tiles per wave

__global__ __launch_bounds__(256) void pq_argmax_gather_kernel(
    const float* __restrict__ vecs,   // [N, M*D]
    const float* __restrict__ cb,     // [M, K, D]
    float* __restrict__ out)          // [N, M*D]
{
    __shared__ float lds_cb[K_CB * CBP];   // codebook slice for subspace m, padded
    __shared__ float lds_h2[K_CB];         // 0.5 * ||c||^2 per codeword

    const int m   = blockIdx.x;
    const int tid = threadIdx.x;

    // ---- Stage codebook slice (16 KB) into LDS; compute 0.5*||c||^2 ----
    {
        const float4* src = (const float4*)(cb + ((size_t)m * K_CB + tid) * D_SUB);
        float4 q0 = src[0], q1 = src[1], q2 = src[2], q3 = src[3];
        float4* dst = (float4*)(lds_cb + tid * CBP);   // tid*80 bytes: 16B aligned
        dst[0] = q0; dst[1] = q1; dst[2] = q2; dst[3] = q3;
        float s = q0.x*q0.x + q0.y*q0.y + q0.z*q0.z + q0.w*q0.w
                + q1.x*q1.x + q1.y*q1.y + q1.z*q1.z + q1.w*q1.w
                + q2.x*q2.x + q2.y*q2.y + q2.z*q2.z + q2.w*q2.w
                + q3.x*q3.x + q3.y*q3.y + q3.z*q3.z + q3.w*q3.w;
        lds_h2[tid] = 0.5f * s;
    }
    __syncthreads();

    const int lane = tid & 31;
    const int wave = tid >> 5;
    const int r    = lane & 15;   // row (A) / codeword-in-group (B) / column (C)
    const int h    = lane >> 4;   // half: selects K-dims {0,1} vs {2,3} within a WMMA chunk

    for (int t = 0; t < TPW; ++t) {
        const int tile = blockIdx.y * (WPB * TPW) + wave * TPW + t;
        const int n0   = tile * 16;

        // ---- Load A: 16 vectors x 16 dims as 4 chunks of 16x4 (ISA A layout) ----
        v2f a[4];
        const float* vrow = vecs + (size_t)(n0 + r) * VDIM + m * D_SUB;
        #pragma unroll
        for (int j = 0; j < 4; ++j)
            a[j] = *(const v2f*)(vrow + 4 * j + 2 * h);

        float bv[8];
        int   bi[8];
        #pragma unroll
        for (int v = 0; v < 8; ++v) { bv[v] = -3.0e38f; bi[v] = 0; }

        // ---- Sweep 16 groups of 16 codewords ----
        #pragma unroll 2
        for (int g = 0; g < 16; ++g) {
            const int cw = g * 16 + r;
            const float negh2 = -lds_h2[cw];
            v8f acc;
            #pragma unroll
            for (int v = 0; v < 8; ++v) acc[v] = negh2;   // C init: score = v.c - 0.5*||c||^2

            #pragma unroll
            for (int j = 0; j < 4; ++j) {
                v2f b = *(const v2f*)(lds_cb + cw * CBP + 4 * j + 2 * h);
                acc = __builtin_amdgcn_wmma_f32_16x16x4_f32(
                        false, a[j], false, b, (short)0, acc, false, false);
            }

            #pragma unroll
            for (int v = 0; v < 8; ++v) {
                if (acc[v] > bv[v]) { bv[v] = acc[v]; bi[v] = cw; }  // strict >: first (lowest g) wins
            }
        }

        // ---- Cross-lane argmax within each 16-lane half (branchless, two phases) ----
        // Phase 1: max of values.  Phase 2: min index among lanes that attained the max.
        // xor offsets 1..8 keep lanes 0-15 (rows 0-7) and 16-31 (rows 8-15) independent.
        #pragma unroll
        for (int v = 0; v < 8; ++v) {
            float val = bv[v];
            #pragma unroll
            for (int off = 1; off < 16; off <<= 1)
                val = fmaxf(val, __shfl_xor(val, off, 32));

            int cand = (bv[v] == val) ? bi[v] : 0x7FFFFFFF;
            #pragma unroll
            for (int off = 1; off < 16; off <<= 1) {
                int oc = __shfl_xor(cand, off, 32);
                cand = (oc < cand) ? oc : cand;
            }
            bi[v] = cand;   // every lane in the half now holds the winner for its row v
        }

        // ---- Select this lane's output row winner (stays within the lane's half) ----
        const int myrow = lane >> 1;      // lanes 0-15 -> rows 0-7, lanes 16-31 -> rows 8-15
        const int hh    = lane & 1;
        const int sel   = myrow & 7;
        int myidx = bi[0];
        #pragma unroll
        for (int v = 1; v < 8; ++v)
            myidx = (sel == v) ? bi[v] : myidx;

        // ---- Gather winning codeword from LDS, store coalesced ----
        {
            const float4* s = (const float4*)(lds_cb + myidx * CBP + hh * 8);
            float4 o0 = s[0];
            float4 o1 = s[1];
            float4* d = (float4*)(out + (size_t)(n0 + myrow) * VDIM + m * D_SUB + hh * 8);
            d[0] = o0;
            d[1] = o1;
        }
    }
}

extern "C" void kernel_launch(void* const* d_in, const int* in_sizes, int n_in,
                              void* d_out, int out_size, void* d_ws, size_t ws_size,
                              hipStream_t stream) {
    (void)in_sizes; (void)n_in; (void)d_ws; (void)ws_size; (void)out_size;
    const float* vecs = (const float*)d_in[0];   // (8192, 1024) f32
    const float* cb   = (const float*)d_in[1];   // (64, 256, 16) f32
    float* out        = (float*)d_out;           // (8192, 1024) f32

    dim3 grid(M_SUB, N_TOTAL / (16 * WPB * TPW));   // (64, 16)
    dim3 block(WPB * 32);                            // 256 threads = 8 waves
    pq_argmax_gather_kernel<<<grid, block, 0, stream>>>(vecs, cb, out);
}